// PLE_28080496181557
// MI455X (gfx1250) — compile-verified
//
#include <hip/hip_runtime.h>

#define NBINS 64

typedef __attribute__((ext_vector_type(4))) float v4f;

// Precompute per-bin tables (1 KB in d_ws), enabling enc = min(max(fma(x,I,B), A), Z):
//   I[k] = 1/(upper-lower)            (removes divide)
//   B[k] = -lower * I[k]              (so v = x*I + B, single v_fma)
//   A[k] = lower clamp bound:  0 interior; -inf for k==0 and k==63 (no left clamp)
//   Z[k] = upper clamp bound:  1 interior and k==0; 0 for k==63
// Interior: x<lower => v<0 -> 0 ; x>=upper => v>=1 -> 1 ; else v.   (matches ref)
// k==0:    no "left" term -> enc = min(v,1).
// k==63:   upper=-1 (negative slope): x>=l => v<=0 -> v ; x<l => v>0 -> clamp to 0.
__global__ void ple_prep_kernel(const float* __restrict__ bins,
                                float* __restrict__ ws) {
    int k = threadIdx.x;               // launched with exactly 64 threads
    float lo = bins[k];
    float up = (k < NBINS - 1) ? bins[k + 1] : -1.0f;
    float inv = 1.0f / (up - lo);
    float* B = ws;
    float* I = ws + NBINS;
    float* A = ws + 2 * NBINS;
    float* Z = ws + 3 * NBINS;
    I[k] = inv;
    B[k] = -lo * inv;
    A[k] = (k == 0 || k == NBINS - 1) ? -__builtin_inff() : 0.0f;
    Z[k] = (k == NBINS - 1) ? 0.0f : 1.0f;
}

// Each wave32 processes 32 x-values -> 32 output rows of 64 floats.
// Lane l owns columns 4c..4c+3 (c = l&15); its 4 table entries live in VGPRs.
// Round r: lanes 0..15 emit row xbase+2r, lanes 16..31 emit row xbase+2r+1
// (x broadcast via ds_bpermute), so each global_store_b128 covers a contiguous
// 512B across the wave, and the store address advances by a constant 512B per
// round (folded into the immediate offset). Stores are TH_NT: 256MB output
// stream > 192MB L2, don't thrash it.
__global__ __launch_bounds__(256) void ple_main_kernel(const float* __restrict__ x,
                                                       const float* __restrict__ ws,
                                                       float* __restrict__ out,
                                                       int n) {
    const int lane  = threadIdx.x & 31;
    const int wave  = blockIdx.x * (blockDim.x >> 5) + (threadIdx.x >> 5);
    const int xbase = wave * 32;
    const int c     = lane & 15;          // column tile owned by this lane
    const int half  = lane >> 4;          // 0 or 1

    // Per-lane register-resident tables (4 x global_load_b128, reused all rounds)
    const v4f Bv = ((const v4f*)(ws))[c];
    const v4f Iv = ((const v4f*)(ws + NBINS))[c];
    const v4f Av = ((const v4f*)(ws + 2 * NBINS))[c];
    const v4f Zv = ((const v4f*)(ws + 3 * NBINS))[c];

    // One coalesced x load per lane; broadcast later via ds_bpermute
    const int xi = xbase + lane;
    const float xm = (xi < n) ? x[xi] : 0.0f;
    const int xbits = __float_as_int(xm);
    const int idx0  = half << 2;          // bpermute byte index base

    // Hoisted store base: round r writes at p + r*128 floats (+512B immediates)
    float* const p = out + (size_t)(xbase + half) * NBINS + (c << 2);

    if (xbase + 32 <= n) {                // whole tile in range: no per-round guards
#pragma unroll
        for (int r = 0; r < 16; ++r) {
            const float xv =
                __int_as_float(__builtin_amdgcn_ds_bpermute(idx0 + r * 8, xbits));
            v4f e;
#pragma unroll
            for (int j = 0; j < 4; ++j) {
                const float v = __builtin_fmaf(xv, Iv[j], Bv[j]);  // (x-l)/(u-l)
                e[j] = __builtin_fminf(__builtin_fmaxf(v, Av[j]), Zv[j]);
            }
            __builtin_nontemporal_store(e, (v4f*)(p + r * 2 * NBINS));
        }
    } else {                              // ragged tail tile
#pragma unroll
        for (int r = 0; r < 16; ++r) {
            const float xv =
                __int_as_float(__builtin_amdgcn_ds_bpermute(idx0 + r * 8, xbits));
            v4f e;
#pragma unroll
            for (int j = 0; j < 4; ++j) {
                const float v = __builtin_fmaf(xv, Iv[j], Bv[j]);
                e[j] = __builtin_fminf(__builtin_fmaxf(v, Av[j]), Zv[j]);
            }
            if (xbase + r * 2 + half < n) {
                __builtin_nontemporal_store(e, (v4f*)(p + r * 2 * NBINS));
            }
        }
    }
}

extern "C" void kernel_launch(void* const* d_in, const int* in_sizes, int n_in,
                              void* d_out, int out_size, void* d_ws, size_t ws_size,
                              hipStream_t stream) {
    const float* x    = (const float*)d_in[0];   // (N,1) f32
    const float* bins = (const float*)d_in[1];   // (64,) f32, sorted
    float* out = (float*)d_out;                  // (N,1,64) f32
    float* ws  = (float*)d_ws;                   // uses 4*64 floats
    const int n = in_sizes[0];

    ple_prep_kernel<<<1, NBINS, 0, stream>>>(bins, ws);

    const int xs_per_block = 256;                // 8 waves * 32 rows each
    const int blocks = (n + xs_per_block - 1) / xs_per_block;
    ple_main_kernel<<<blocks, 256, 0, stream>>>(x, ws, out, n);
}